// Blur2_1365799600519
// MI455X (gfx1250) — compile-verified
//
#include <hip/hip_runtime.h>

typedef __attribute__((ext_vector_type(2))) float v2f;
typedef __attribute__((ext_vector_type(8))) float v8f;

#define H_IMG 256
#define W_IMG 256
#define TILE 16
#define RAW_W 21   // 19 halo cols + 2 pad (bank-conflict dodge + async overrun slot)
#define RAW_ROWS 19
#define HB_W 17    // 16 cols + pad
#define HB_ROWS 20 // 19 rows + 1 zero pad row (K padded to 20)

#if defined(__gfx1250__) && __has_builtin(__builtin_amdgcn_global_load_async_to_lds_b32)
#define USE_ASYNC_LDS 1
#else
#define USE_ASYNC_LDS 0
#endif

#if USE_ASYNC_LDS
typedef __attribute__((address_space(1))) int gas_int;  // global (AS1)
typedef __attribute__((address_space(3))) int las_int;  // LDS (AS3)

__device__ __forceinline__ void wait_async_zero() {
#if __has_builtin(__builtin_amdgcn_s_wait_asynccnt)
    __builtin_amdgcn_s_wait_asynccnt(0);
#else
    asm volatile("s_wait_asynccnt 0x0" ::: "memory");
#endif
}
#endif

// Banded vertical-pass weight: Wv[m][r] = k1d[r-m], k1d = [1,3,3,1]/4
__device__ __forceinline__ float band_w(int m, int r) {
    int d = r - m;
    float w = 0.0f;
    if (d == 0 || d == 3) w = 0.25f;
    if (d == 1 || d == 2) w = 0.75f;
    return w;
}

__global__ __launch_bounds__(256) void blur2_wmma_kernel(const float* __restrict__ in,
                                                         float* __restrict__ out) {
    const int lane = threadIdx.x & 31;
    const int wid  = threadIdx.x >> 5;   // wave32: 8 waves per block

    __shared__ float raw[8][RAW_ROWS * RAW_W]; // per-wave raw halo tile
    __shared__ float hb [8][HB_ROWS * HB_W];   // per-wave horizontally-filtered 20x16

    // global wave id -> (image, 16x16 tile); 256 tiles per 256x256 image
    const int  gw   = blockIdx.x * 8 + wid;
    const int  img  = gw >> 8;
    const int  tidx = gw & 255;
    const int  ty   = tidx >> 4;
    const int  tx   = tidx & 15;
    const int  y0   = ty * TILE;
    const int  x0   = tx * TILE;
    const long base = (long)img * (H_IMG * W_IMG);

    // interior tiles: whole 19x20 halo window is in-bounds (no zero padding needed)
    const bool interior = (ty >= 1) & (ty <= 14) & (tx >= 1) & (tx <= 14);

    // ---- Phase 1: fill 19x19(+pad) halo tile in LDS ----
#if USE_ASYNC_LDS
    if (interior) {
        // Stream rows straight into LDS via GLOBAL_LOAD_ASYNC_TO_LDS_B32:
        // no VGPR round trip, tracked on ASYNCcnt.
        const float* src = in + base + (long)(y0 - 2) * W_IMG + (x0 - 2);
        for (int idx = lane; idx < RAW_ROWS * 20; idx += 32) {
            int r = idx / 20;
            int c = idx % 20;  // c==19 is valid memory; lands in LDS pad slot
            gas_int* gptr = (gas_int*)(src + (long)r * W_IMG + c);
            las_int* lptr = (las_int*)&raw[wid][r * RAW_W + c];
            __builtin_amdgcn_global_load_async_to_lds_b32(gptr, lptr, 0, 0);
        }
        wait_async_zero();   // barrier does not drain ASYNCcnt; wave reads own region next
    } else
#endif
    {
        for (int idx = lane; idx < RAW_ROWS * 20; idx += 32) {
            int r = idx / 20;
            int c = idx % 20;
            if (c < 19) {
                int gy = y0 + r - 2;
                int gx = x0 + c - 2;
                float v = 0.0f;
                if (gy >= 0 && gy < H_IMG && gx >= 0 && gx < W_IMG)
                    v = in[base + (long)gy * W_IMG + gx];
                raw[wid][r * RAW_W + c] = v;
            }
        }
    }
    __syncthreads();

    // ---- Phase 2: horizontal 4-tap [0.25,0.75,0.75,0.25] (VALU) ----
    for (int idx = lane; idx < RAW_ROWS * TILE; idx += 32) {
        int r = idx >> 4;
        int c = idx & 15;
        const float* rp = &raw[wid][r * RAW_W + c];
        hb[wid][r * HB_W + c] =
            0.25f * rp[0] + 0.75f * rp[1] + 0.75f * rp[2] + 0.25f * rp[3];
    }
    if (lane < 16) hb[wid][19 * HB_W + lane] = 0.0f; // zero K-pad row
    __syncthreads();

    // ---- Phase 3: vertical 4-tap as banded matmul, 5x V_WMMA_F32_16X16X4_F32 ----
    // A 16x4 layout: lanes 0-15 hold (K0 in v0, K1 in v1), lanes 16-31 hold (K2, K3).
    // B 4x16 layout mirrors A: lanes 0-15 rows {K0,K1}, lanes 16-31 rows {K2,K3}, N = lane&15.
    const int m    = lane & 15;
    const int koff = (lane >> 4) << 1;
    v8f acc = {};
#pragma unroll
    for (int c = 0; c < 5; ++c) {
        int k0 = 4 * c + koff;
        v2f a, b;
        a.x = band_w(m, k0);
        a.y = band_w(m, k0 + 1);
        b.x = hb[wid][k0 * HB_W + m];
        b.y = hb[wid][(k0 + 1) * HB_W + m];
        acc = __builtin_amdgcn_wmma_f32_16x16x4_f32(false, a, false, b,
                                                    (short)0, acc, false, false);
    }

    // ---- Phase 4: store D (C/D layout: VGPR i -> M=i / M=i+8 per lane half) ----
    const int col   = x0 + m;
    const int rbase = y0 + (lane >> 4) * 8;
#pragma unroll
    for (int i = 0; i < 8; ++i) {
        __builtin_nontemporal_store(acc[i], &out[base + (long)(rbase + i) * W_IMG + col]);
    }
}

extern "C" void kernel_launch(void* const* d_in, const int* in_sizes, int n_in,
                              void* d_out, int out_size, void* d_ws, size_t ws_size,
                              hipStream_t stream) {
    const float* in  = (const float*)d_in[0];
    float*       out = (float*)d_out;
    // 8*128 images * (256/16)^2 tiles = 262144 waves; 8 waves per 256-thread block
    const int blocks = (8 * 128 * (H_IMG / TILE) * (W_IMG / TILE)) / 8;
    blur2_wmma_kernel<<<blocks, 256, 0, stream>>>(in, out);
}